// GAT_1726576853727
// MI455X (gfx1250) — compile-verified
//
#include <hip/hip_runtime.h>
#include <hip/hip_bf16.h>

typedef __attribute__((ext_vector_type(2))) float v2f;
typedef __attribute__((ext_vector_type(8))) float v8f;

#define GAT_H    4
#define GAT_B    32
#define GAT_N    14
#define GAT_IF   1024
#define GAT_OF   1024
#define GAT_F    (GAT_N * GAT_OF)     /* 14336 */
#define ALPHA    0.2f
#define NEGV     (-9.0e15f)
#define XS_STRIDE 1028                /* 1024 + 4 floats: 4-bank row skew, conflict-free */

static __device__ __forceinline__ v8f wmma_f32_16x16x4(v2f a, v2f b, v8f c) {
  // 8 args: (neg_a, A, neg_b, B, c_mod, C, reuse_a, reuse_b)
  return __builtin_amdgcn_wmma_f32_16x16x4_f32(false, a, false, b, (short)0, c,
                                               false, false);
}

// ---------------------------------------------------------------------------
// Kernel 0: out_pre[b][o] = sum_h fc_b[h][o]   (re-run every call => graph-safe)
// ---------------------------------------------------------------------------
__global__ __launch_bounds__(256) void k_init(const float* __restrict__ fcb,
                                              float* __restrict__ outp) {
  int i = blockIdx.x * 256 + threadIdx.x;          // 32*1024 elements
  if (i < GAT_B * GAT_OF) {
    int o = i & (GAT_OF - 1);
    outp[i] = fcb[o] + fcb[GAT_OF + o] + fcb[2 * GAT_OF + o] + fcb[3 * GAT_OF + o];
  }
}

// ---------------------------------------------------------------------------
// Kernel 1: Wh[h,b] = x[b] (14x1024, LDS) @ W[h,b] (1024x1024, streamed once)
// grid = (H*B, OF/128); block = 256 (8 waves, one 16x16 WMMA tile each)
// ---------------------------------------------------------------------------
__global__ __launch_bounds__(256) void k_wh(const float* __restrict__ x,
                                            const float* __restrict__ W,
                                            float* __restrict__ Wh) {
  extern __shared__ float xs[];                    // 16 * XS_STRIDE floats
  const int hb   = blockIdx.x;                     // h*32 + b
  const int b    = hb & 31;
  const int lane = threadIdx.x & 31;
  const int wave = threadIdx.x >> 5;

  // Stage x[b] into LDS, zero rows 14..15 (M padding for the 16x16 tile).
  for (int i = threadIdx.x; i < 16 * 256; i += 256) {
    const int row = i >> 8;
    const int c4  = (i & 255) << 2;
    float4 v = make_float4(0.f, 0.f, 0.f, 0.f);
    if (row < GAT_N)
      v = *(const float4*)(x + ((size_t)b * GAT_N + row) * GAT_IF + c4);
    *(float4*)(&xs[row * XS_STRIDE + c4]) = v;
  }
  __syncthreads();

  const int n0    = blockIdx.y * 128 + wave * 16;
  const int nl    = lane & 15;
  const int krow  = (lane >> 4) << 1;              // 0 (lanes 0-15) or 2 (16-31)
  const int abase = nl * XS_STRIDE + krow;
  const float* wcol = W + (size_t)hb * GAT_IF * GAT_OF + n0 + nl;

  v8f acc = {};
#pragma unroll 8
  for (int k = 0; k < GAT_IF; k += 4) {
    v2f av;
    av.x = xs[abase + k];                          // A[m][k+krow]
    av.y = xs[abase + k + 1];                      // A[m][k+krow+1]
    v2f bv;
    bv.x = wcol[(size_t)(k + krow) * GAT_OF];      // B[k+krow][n]
    bv.y = wcol[(size_t)(k + krow + 1) * GAT_OF];
    acc = wmma_f32_16x16x4(av, bv, acc);
  }

  float* o = Wh + (size_t)hb * GAT_N * GAT_OF;
  const int mb = (lane < 16) ? 0 : 8;
#pragma unroll
  for (int r = 0; r < 8; ++r) {
    const int m = mb + r;
    if (m < GAT_N) o[(size_t)m * GAT_OF + n0 + nl] = acc[r];
  }
}

// ---------------------------------------------------------------------------
// Kernel 2: attention per (h,b): Wh1/Wh2 dots, masked leaky-relu,
// softmax over n (axis=2), h_prime = att @ Wh.  grid = H*B, block = 256.
// ---------------------------------------------------------------------------
__global__ __launch_bounds__(256) void k_att(const float* __restrict__ Wh,
                                             const float* __restrict__ adj,
                                             const float* __restrict__ a,
                                             float* __restrict__ hp) {
  const int hb = blockIdx.x;
  const int b  = hb & 31;
  const int t  = threadIdx.x;
  const float* whp = Wh + (size_t)hb * GAT_N * GAT_OF;
  const float* a1  = a + (size_t)hb * (2 * GAT_OF);
  const float* a2  = a1 + GAT_OF;

  __shared__ float red[256];
  __shared__ float wh1[GAT_N];
  __shared__ float wh2[GAT_N];
  __shared__ float att[GAT_N * 16];                // att[n][m], stride 16

  for (int n = 0; n < GAT_N; ++n) {
    float s1 = 0.f, s2 = 0.f;
    for (int o = t; o < GAT_OF; o += 256) {
      float v = whp[n * GAT_OF + o];
      s1 += v * a1[o];
      s2 += v * a2[o];
    }
    red[t] = s1; __syncthreads();
    for (int s = 128; s > 0; s >>= 1) { if (t < s) red[t] += red[t + s]; __syncthreads(); }
    if (t == 0) wh1[n] = red[0];
    __syncthreads();
    red[t] = s2; __syncthreads();
    for (int s = 128; s > 0; s >>= 1) { if (t < s) red[t] += red[t + s]; __syncthreads(); }
    if (t == 0) wh2[n] = red[0];
    __syncthreads();
  }

  // Column-wise softmax (axis=2 of (H,B,N,N) => normalize over n per column m).
  if (t < GAT_N) {
    const int m = t;
    float col[GAT_N];
    float mx = -__builtin_inff();
    for (int n = 0; n < GAT_N; ++n) {
      float e = wh1[n] + wh2[m];
      e = (e > 0.f) ? e : ALPHA * e;               // leaky relu
      float ad = adj[(size_t)b * GAT_N * GAT_N + n * GAT_N + m];
      e = (ad > 0.f) ? e : NEGV;
      col[n] = e;
      mx = fmaxf(mx, e);
    }
    float sum = 0.f;
    for (int n = 0; n < GAT_N; ++n) { float ex = __expf(col[n] - mx); col[n] = ex; sum += ex; }
    float inv = 1.f / sum;
    for (int n = 0; n < GAT_N; ++n) att[n * 16 + m] = col[n] * inv;
  }
  __syncthreads();

  // h_prime[n][o] = sum_m att[n][m] * Wh[m][o]
  float* hpp = hp + (size_t)hb * GAT_N * GAT_OF;
  for (int o = t; o < GAT_OF; o += 256) {
    float wcol[GAT_N];
#pragma unroll
    for (int m = 0; m < GAT_N; ++m) wcol[m] = whp[m * GAT_OF + o];
#pragma unroll
    for (int n = 0; n < GAT_N; ++n) {
      float acc = 0.f;
#pragma unroll
      for (int m = 0; m < GAT_N; ++m) acc += att[n * 16 + m] * wcol[m];
      hpp[n * GAT_OF + o] = acc;
    }
  }
}

// ---------------------------------------------------------------------------
// Kernel 3: out_pre[b,o] += sum_h sum_f hp[h,b,f] * fc_w[h,o,f]
// Split-K WMMA GEMM: grid = (OF/64, H * 7 k-chunks of 2048), block = 256.
// Waves: m_tile = wave&1 (rows 0-15 / 16-31), n_sub = wave>>1.
// ---------------------------------------------------------------------------
__global__ __launch_bounds__(256) void k_fc(const float* __restrict__ hp,
                                            const float* __restrict__ fcw,
                                            float* __restrict__ outp) {
  const int lane = threadIdx.x & 31;
  const int wave = threadIdx.x >> 5;
  const int m0   = (wave & 1) * 16;
  const int n0   = blockIdx.x * 64 + (wave >> 1) * 16;
  const int h    = blockIdx.y & 3;
  const int k0   = (blockIdx.y >> 2) * 2048;
  const int nl   = lane & 15;
  const int krow = (lane >> 4) << 1;

  const float* A  = hp  + (size_t)h * GAT_B  * GAT_F + (size_t)(m0 + nl) * GAT_F + k0 + krow;
  const float* Bm = fcw + (size_t)h * GAT_OF * GAT_F + (size_t)(n0 + nl) * GAT_F + k0 + krow;

  v8f acc = {};
#pragma unroll 8
  for (int k = 0; k < 2048; k += 4) {
    v2f av; av.x = A[k];  av.y = A[k + 1];         // contiguous -> b64 load
    v2f bv; bv.x = Bm[k]; bv.y = Bm[k + 1];
    acc = wmma_f32_16x16x4(av, bv, acc);
  }

  const int mb = (lane < 16) ? 0 : 8;
#pragma unroll
  for (int r = 0; r < 8; ++r)
    atomicAdd(&outp[(size_t)(m0 + mb + r) * GAT_OF + n0 + nl], acc[r]);
}

// ---------------------------------------------------------------------------
// Kernel 4: log_softmax per row b over 1024 cols. grid = 32, block = 256.
// ---------------------------------------------------------------------------
__global__ __launch_bounds__(256) void k_lsm(const float* __restrict__ outp,
                                             float* __restrict__ out) {
  const int b = blockIdx.x;
  const int t = threadIdx.x;
  __shared__ float red[256];
  const float* row = outp + (size_t)b * GAT_OF;

  float mx = -__builtin_inff();
  for (int o = t; o < GAT_OF; o += 256) mx = fmaxf(mx, row[o]);
  red[t] = mx; __syncthreads();
  for (int s = 128; s > 0; s >>= 1) { if (t < s) red[t] = fmaxf(red[t], red[t + s]); __syncthreads(); }
  mx = red[0]; __syncthreads();

  float sum = 0.f;
  for (int o = t; o < GAT_OF; o += 256) sum += __expf(row[o] - mx);
  red[t] = sum; __syncthreads();
  for (int s = 128; s > 0; s >>= 1) { if (t < s) red[t] += red[t + s]; __syncthreads(); }
  const float lse = mx + __logf(red[0]);
  __syncthreads();

  for (int o = t; o < GAT_OF; o += 256) out[(size_t)b * GAT_OF + o] = row[o] - lse;
}

// ---------------------------------------------------------------------------
extern "C" void kernel_launch(void* const* d_in, const int* in_sizes, int n_in,
                              void* d_out, int out_size, void* d_ws, size_t ws_size,
                              hipStream_t stream) {
  const float* x   = (const float*)d_in[0];  // (B,N,IF)
  const float* adj = (const float*)d_in[1];  // (B,N,N)
  const float* W   = (const float*)d_in[2];  // (H,B,IF,OF)
  const float* a   = (const float*)d_in[3];  // (H,B,2*OF,1)
  const float* fcw = (const float*)d_in[4];  // (H,OF,N*OF)
  const float* fcb = (const float*)d_in[5];  // (H,OF)
  float* out = (float*)d_out;                // (B,OF)

  const size_t WH_ELEMS = (size_t)GAT_H * GAT_B * GAT_N * GAT_OF;  // 1,835,008
  float* Wh      = (float*)d_ws;
  float* hpbuf   = Wh + WH_ELEMS;
  float* out_pre = hpbuf + WH_ELEMS;

  (void)in_sizes; (void)n_in; (void)out_size; (void)ws_size;

  k_init<<<(GAT_B * GAT_OF + 255) / 256, 256, 0, stream>>>(fcb, out_pre);

  const size_t lds_bytes = (size_t)16 * XS_STRIDE * sizeof(float);  // 65,792 B
  k_wh<<<dim3(GAT_H * GAT_B, GAT_OF / 128), 256, lds_bytes, stream>>>(x, W, Wh);

  k_att<<<GAT_H * GAT_B, 256, 0, stream>>>(Wh, adj, a, hpbuf);

  k_fc<<<dim3(GAT_OF / 64, GAT_H * (GAT_F / 2048)), 256, 0, stream>>>(hpbuf, fcw, out_pre);

  k_lsm<<<GAT_B, 256, 0, stream>>>(out_pre, out);
}